// SSIM_67224828117331
// MI455X (gfx1250) — compile-verified
//
#include <hip/hip_runtime.h>

// CDNA5 (gfx1250) SSIM via banded-matrix V_WMMA_F32_16X16X4_F32.
// Separable 11-tap Gaussian blur as band matmuls on 16x16 tiles, with a
// ping-pong LDS intermediate and software-pipelined horizontal/vertical passes.

typedef float v2f __attribute__((ext_vector_type(2)));
typedef float v8f __attribute__((ext_vector_type(8)));

#define IMG_H 512
#define IMG_W 512
#define N_TILES 49152            // 16*3 images * 32*32 tiles
#define WAVES_PER_BLOCK 8
#define N_BLOCKS (N_TILES / WAVES_PER_BLOCK)   // 6144
#define LDS_STRIDE 18            // row stride (dwords): conflict-free stores & loads

// Normalized 11-tap Gaussian, sigma = 1.0 (matches reference numpy window)
__device__ __constant__ float GW[11] = {
    1.4867196e-06f, 1.3383022e-04f, 4.4318484e-03f, 5.3990967e-02f,
    2.4197072e-01f, 3.9894228e-01f, 2.4197072e-01f, 5.3990967e-02f,
    4.4318484e-03f, 1.3383022e-04f, 1.4867196e-06f
};

__global__ __launch_bounds__(WAVES_PER_BLOCK * 32)
void ssim_wmma_kernel(const float* __restrict__ pred,
                      const float* __restrict__ targ,
                      float* __restrict__ partials)
{
    __shared__ float Hbuf[WAVES_PER_BLOCK][2][32 * LDS_STRIDE];
    __shared__ float waveSum[WAVES_PER_BLOCK];

    const int lane = threadIdx.x & 31;
    const int wv   = threadIdx.x >> 5;
    const int tile = blockIdx.x * WAVES_PER_BLOCK + wv;   // exactly N_TILES waves

    const int img  = tile >> 10;          // / (32*32)
    const int tIdx = tile & 1023;
    const int r0 = (tIdx >> 5) << 4;
    const int c0 = (tIdx & 31) << 4;

    const float* __restrict__ pimg = pred + (size_t)img * (IMG_H * IMG_W);
    const float* __restrict__ timg = targ + (size_t)img * (IMG_H * IMG_W);

    const int colq  = lane & 15;            // N (or M) index owned by this lane
    const int khalf = (lane >> 4) << 1;     // K sub-offset: 0 or 2
    const int mrow  = lane & 15;            // A-matrix row owned by this lane

    // Band-matrix chunks: value = w(k - c). Identical per-lane data for the
    // horizontal B-role (4x16) and vertical A-role (16x4).
    v2f bw[7];
#pragma unroll
    for (int t = 0; t < 7; ++t) {
        const int k0 = 4 * t + khalf;
        const int d0 = k0 - colq;
        const int d1 = d0 + 1;
        bw[t].x = (d0 >= 0 && d0 <= 10) ? GW[d0] : 0.0f;
        bw[t].y = (d1 >= 0 && d1 <= 10) ? GW[d1] : 0.0f;
    }

    // ---- load 32x28 input patch in WMMA A-layout ----
    v2f ax[2][7], ay[2][7];

    auto load_checked = [&]() {
#pragma unroll
        for (int rt = 0; rt < 2; ++rt) {
            const int grow = r0 - 5 + 16 * rt + mrow;
            const bool rok = (grow >= 0) && (grow < IMG_H);
            const long ro = (long)grow * IMG_W;
#pragma unroll
            for (int t = 0; t < 7; ++t) {
                const int gc = c0 - 5 + 4 * t + khalf;
                const bool ok0 = rok && (gc >= 0) && (gc < IMG_W);
                const bool ok1 = rok && (gc + 1 >= 0) && (gc + 1 < IMG_W);
                ax[rt][t].x = ok0 ? pimg[ro + gc]     : 0.0f;
                ax[rt][t].y = ok1 ? pimg[ro + gc + 1] : 0.0f;
                ay[rt][t].x = ok0 ? timg[ro + gc]     : 0.0f;
                ay[rt][t].y = ok1 ? timg[ro + gc + 1] : 0.0f;
            }
        }
    };
    auto load_fast = [&]() {
#pragma unroll
        for (int rt = 0; rt < 2; ++rt) {
            const long ro = (long)(r0 - 5 + 16 * rt + mrow) * IMG_W;
#pragma unroll
            for (int t = 0; t < 7; ++t) {
                const int gc = c0 - 5 + 4 * t + khalf;
                ax[rt][t].x = pimg[ro + gc];
                ax[rt][t].y = pimg[ro + gc + 1];
                ay[rt][t].x = timg[ro + gc];
                ay[rt][t].y = timg[ro + gc + 1];
            }
        }
    };
    // Border condition is wave-uniform; force a scalar branch so EXEC stays
    // all-ones (WMMA requirement) and the 88% interior tiles skip predication.
    const int interior = (r0 != 0) & (r0 != 496) & (c0 != 0) & (c0 != 496);
    if (__builtin_amdgcn_readfirstlane(interior)) load_fast();
    else                                          load_checked();

    // per-map A chunk: x, y, x*x, y*y, x*y
    auto make_a = [&](int q, v2f x, v2f y) -> v2f {
        if (q == 0) return x;
        if (q == 1) return y;
        if (q == 2) return x * x;
        if (q == 3) return y * y;
        return x * y;
    };

    // Horizontal pass: two 16-row tiles of D(16x16) = P(16x28) x Band(28x16),
    // D tiles stored to LDS buffer Hdst.
    auto horiz_pass = [&](int q, float* Hdst) {
#pragma unroll
        for (int rt = 0; rt < 2; ++rt) {
            v8f acc = {};
#pragma unroll
            for (int t = 0; t < 7; ++t) {
                const v2f a = make_a(q, ax[rt][t], ay[rt][t]);
                acc = __builtin_amdgcn_wmma_f32_16x16x4_f32(
                        false, a, false, bw[t], (short)0, acc, false, false);
            }
            const int rbase = ((lane >> 4) << 3) + 16 * rt;
#pragma unroll
            for (int d = 0; d < 8; ++d)
                Hdst[(rbase + d) * LDS_STRIDE + colq] = acc[d];
        }
    };

    // ---- software-pipelined map loop with ping-pong H buffers ----
    v8f res[5];
    horiz_pass(0, Hbuf[wv][0]);
#pragma unroll
    for (int q = 0; q < 5; ++q) {
        float* Hsrc = Hbuf[wv][q & 1];
        // 1) issue all vertical B-chunk loads for map q
        v2f b[7];
#pragma unroll
        for (int t = 0; t < 7; ++t) {
            const int k0 = 4 * t + khalf;
            b[t].x = Hsrc[(k0    ) * LDS_STRIDE + colq];
            b[t].y = Hsrc[(k0 + 1) * LDS_STRIDE + colq];
        }
        // 2) independent work under the load latency: horizontal pass of q+1
        if (q < 4) horiz_pass(q + 1, Hbuf[wv][(q + 1) & 1]);
        // 3) vertical chain: D = Band(16x28) x H(28x16)
        v8f accv = {};
#pragma unroll
        for (int t = 0; t < 7; ++t)
            accv = __builtin_amdgcn_wmma_f32_16x16x4_f32(
                    false, bw[t], false, b[t], (short)0, accv, false, false);
        res[q] = accv;
    }

    // ---- SSIM map (8 pixels per lane) + tile sum ----
    const float C1 = 1.0e-4f;   // (0.01 * 1.0)^2
    const float C2 = 9.0e-4f;   // (0.03 * 1.0)^2
    float s = 0.0f;
#pragma unroll
    for (int d = 0; d < 8; ++d) {
        const float mx  = res[0][d], my = res[1][d];
        const float sx  = res[2][d] - mx * mx;
        const float sy  = res[3][d] - my * my;
        const float sxy = res[4][d] - mx * my;
        const float num = (2.0f * mx * my + C1) * (2.0f * sxy + C2);
        const float den = (mx * mx + my * my + C1) * (sx * sx + sy * sy + C2);
        s += num / den;
    }

    // wave32 reduction, then per-block partial (deterministic, no atomics)
#pragma unroll
    for (int off = 16; off >= 1; off >>= 1)
        s += __shfl_xor(s, off, 32);
    if (lane == 0) waveSum[wv] = s;
    __syncthreads();
    if (threadIdx.x == 0) {
        float bs = 0.0f;
#pragma unroll
        for (int w = 0; w < WAVES_PER_BLOCK; ++w) bs += waveSum[w];
        partials[blockIdx.x] = bs;
    }
}

__global__ __launch_bounds__(256)
void ssim_reduce_kernel(const float* __restrict__ partials,
                        float* __restrict__ out, float invN)
{
    __shared__ float sm[256];
    float s = 0.0f;
    for (int i = threadIdx.x; i < N_BLOCKS; i += 256)
        s += partials[i];
    sm[threadIdx.x] = s;
    __syncthreads();
    for (int off = 128; off > 0; off >>= 1) {
        if (threadIdx.x < off) sm[threadIdx.x] += sm[threadIdx.x + off];
        __syncthreads();
    }
    if (threadIdx.x == 0) out[0] = sm[0] * invN;
}

extern "C" void kernel_launch(void* const* d_in, const int* in_sizes, int n_in,
                              void* d_out, int out_size, void* d_ws, size_t ws_size,
                              hipStream_t stream) {
    const float* pred = (const float*)d_in[0];
    const float* targ = (const float*)d_in[1];
    float* out      = (float*)d_out;
    float* partials = (float*)d_ws;   // N_BLOCKS floats = 24 KB

    ssim_wmma_kernel<<<N_BLOCKS, WAVES_PER_BLOCK * 32, 0, stream>>>(pred, targ, partials);

    const float invN = 1.0f / (16.0f * 3.0f * 512.0f * 512.0f);
    ssim_reduce_kernel<<<1, 256, 0, stream>>>(partials, out, invN);
}